// GNNModel_19361712570396
// MI455X (gfx1250) — compile-verified
//
#include <hip/hip_runtime.h>

#define NN 50000

typedef float v8f __attribute__((ext_vector_type(8)));
typedef float v2f __attribute__((ext_vector_type(2)));

// ---------------------------------------------------------------- utilities
__global__ void fill_f32(float* __restrict__ p, float v, int n) {
    int t = blockIdx.x * blockDim.x + threadIdx.x;
    if (t < n) p[t] = v;
}

__global__ void deg_scatter(const int* __restrict__ col, float* __restrict__ deg, int E) {
    int t = blockIdx.x * blockDim.x + threadIdx.x;
    if (t < E) atomicAdd(&deg[col[t]], 1.0f);
}

__global__ void deg_to_dis(float* __restrict__ d, int n) {
    int t = blockIdx.x * blockDim.x + threadIdx.x;
    if (t < n) {
        float v = d[t];
        d[t] = (v > 0.0f) ? rsqrtf(fmaxf(v, 1.0f)) : 0.0f;
    }
}

__global__ void edge_norm(const int* __restrict__ row, const int* __restrict__ col,
                          const float* __restrict__ dis, float* __restrict__ norm, int E) {
    int t = blockIdx.x * blockDim.x + threadIdx.x;
    if (t < E) norm[t] = dis[row[t]] * dis[col[t]];
}

// -------------------------------------------------- dense transform: WMMA f32
// One wave computes one 16x16 tile of H = A(MxK) * W(KxF).  M = 50000 = 3125*16
// exactly, so no M-tail.  K/F/RELU are compile-time: all in-loop bounds checks
// fold away, the K-loop fully unrolls, and A-fragments load as b64.
//
// f32 16x16x4 fragment layouts (ISA 7.12.2):
//   A 16x4 : lanes 0-15 -> M=lane (K=0,1 in v0,v1), lanes 16-31 -> K=2,3
//   B 4x16 : lanes 0-15 -> N=lane (K=0,1 in v0,v1), lanes 16-31 -> K=2,3
//   C/D    : VGPR j, lanes 0-15 -> row j, lanes 16-31 -> row j+8
template <int K, int F, bool RELU>
__global__ void gemm_wmma_f32(const float* __restrict__ A, const float* __restrict__ W,
                              float* __restrict__ H, int totalTiles) {
    constexpr int tilesN = F / 16;
    int wave = blockIdx.x * (blockDim.x >> 5) + (threadIdx.x >> 5);
    if (wave >= totalTiles) return;           // whole wave exits: EXEC full for WMMA
    int lane = threadIdx.x & 31;
    int half = lane >> 4;                     // 0: K pair {0,1}, 1: K pair {2,3}
    int lm   = lane & 15;
    int tm = wave / tilesN;
    int tn = wave - tm * tilesN;
    const float* __restrict__ Arow = A + (size_t)(tm * 16 + lm) * K;
    int bcol = tn * 16 + lm;

    v8f c = {};
    constexpr int KM = K & ~3;                // main body (multiple of 4)
#pragma unroll
    for (int k0 = 0; k0 < KM; k0 += 4) {
        int ka = k0 + 2 * half;
        float a0, a1;
        if constexpr ((K & 1) == 0) {         // even K: 8B-aligned pair -> b64 load
            float2 a2 = *(const float2*)(Arow + ka);
            a0 = a2.x; a1 = a2.y;
        } else {                              // odd K (29): keep dword loads
            a0 = Arow[ka]; a1 = Arow[ka + 1];
        }
        if constexpr (RELU) { a0 = fmaxf(a0, 0.0f); a1 = fmaxf(a1, 0.0f); }
        float b0 = W[(size_t)ka * F + bcol];
        float b1 = W[(size_t)(ka + 1) * F + bcol];
        v2f av; av.x = a0; av.y = a1;
        v2f bv; bv.x = b0; bv.y = b1;
        c = __builtin_amdgcn_wmma_f32_16x16x4_f32(false, av, false, bv,
                                                  (short)0, c, false, false);
    }
    if constexpr (KM < K) {                   // K tail, branch-free (clamp + select)
        int ka = KM + 2 * half;
        int i0 = (ka     < K) ? ka       : (K - 1);
        int i1 = (ka + 1 < K) ? (ka + 1) : (K - 1);
        float a0 = Arow[i0];
        float a1 = Arow[i1];
        float b0 = W[(size_t)i0 * F + bcol];
        float b1 = W[(size_t)i1 * F + bcol];
        if (ka     >= K) { a0 = 0.0f; b0 = 0.0f; }
        if (ka + 1 >= K) { a1 = 0.0f; b1 = 0.0f; }
        if constexpr (RELU) { a0 = fmaxf(a0, 0.0f); a1 = fmaxf(a1, 0.0f); }
        v2f av; av.x = a0; av.y = a1;
        v2f bv; bv.x = b0; bv.y = b1;
        c = __builtin_amdgcn_wmma_f32_16x16x4_f32(false, av, false, bv,
                                                  (short)0, c, false, false);
    }

    int col = tn * 16 + lm;
#pragma unroll
    for (int j = 0; j < 8; ++j) {
        int r = tm * 16 + j + 8 * half;
        H[(size_t)r * F + col] = c[j];
    }
}

// -------------------------------------------------- aggregation (scatter-add)
// out[i,f] = b[f] + h[i,f]*dis[i]^2   (bias + self-loop message)
template <int F>
__global__ void agg_init(const float* __restrict__ h, const float* __restrict__ dis,
                         const float* __restrict__ bias, float* __restrict__ out) {
    int t = blockIdx.x * blockDim.x + threadIdx.x;
    if (t >= NN * F) return;
    int i = t / F;
    int f = t - i * F;
    float d = dis[i];
    out[t] = bias[f] + h[t] * d * d;
}

// one thread per (edge, feature); lanes run along f -> coalesced gather+atomic
template <int F>
__global__ void agg_scatter(const float* __restrict__ h, const int* __restrict__ row,
                            const int* __restrict__ col, const float* __restrict__ norm,
                            float* __restrict__ out, int E) {
    long long t = (long long)blockIdx.x * blockDim.x + threadIdx.x;
    if (t >= (long long)E * F) return;
    int e = (int)(t / F);
    int f = (int)(t - (long long)e * F);
    float m = h[(size_t)row[e] * F + f] * norm[e];
    atomicAdd(&out[(size_t)col[e] * F + f], m);
}

// ---------------------------------------------------------------- launcher
static inline int ceil_div(long long a, long long b) { return (int)((a + b - 1) / b); }

extern "C" void kernel_launch(void* const* d_in, const int* in_sizes, int n_in,
                              void* d_out, int out_size, void* d_ws, size_t ws_size,
                              hipStream_t stream) {
    const float* x  = (const float*)d_in[0];
    const int*   ei = (const int*)d_in[1];
    const float* W1 = (const float*)d_in[2];
    const float* b1 = (const float*)d_in[3];
    const float* W2 = (const float*)d_in[4];
    const float* b2 = (const float*)d_in[5];
    const float* W3 = (const float*)d_in[6];
    const float* b3 = (const float*)d_in[7];
    const float* W4 = (const float*)d_in[8];
    const float* b4 = (const float*)d_in[9];

    const int E = in_sizes[1] / 2;           // 800000
    const int* row = ei;                     // edge_index[0]
    const int* col = ei + E;                 // edge_index[1]

    // workspace carve-up (offsets kept 1KB-aligned)
    float* dis  = (float*)d_ws;                                   // NN
    float* norm = dis  + ((NN + 255) & ~255);                     // E
    float* bufA = norm + ((E  + 255) & ~255);                     // NN*128
    float* bufB = bufA + (size_t)NN * 128;                        // NN*128

    const int TB = 256;

    // ---- graph norm --------------------------------------------------------
    fill_f32<<<ceil_div(NN, TB), TB, 0, stream>>>(dis, 1.0f, NN);            // self-loop deg
    deg_scatter<<<ceil_div(E, TB), TB, 0, stream>>>(col, dis, E);
    deg_to_dis<<<ceil_div(NN, TB), TB, 0, stream>>>(dis, NN);
    edge_norm<<<ceil_div(E, TB), TB, 0, stream>>>(row, col, dis, norm, E);

    const int TILES_M = NN / 16;             // 3125, exact
    const int GB = 128;                      // 4 waves / block

    // ---- layer 1: 29 -> 96 (ReLU folded into next GEMM's A load) ----------
    {
        int total = TILES_M * (96 / 16);
        gemm_wmma_f32<29, 96, false><<<ceil_div(total, GB / 32), GB, 0, stream>>>(x, W1, bufA, total);
    }
    agg_init<96><<<ceil_div((long long)NN * 96, TB), TB, 0, stream>>>(bufA, dis, b1, bufB);
    agg_scatter<96><<<ceil_div((long long)E * 96, TB), TB, 0, stream>>>(bufA, row, col, norm, bufB, E);

    // ---- layer 2: 96 -> 128 ------------------------------------------------
    {
        int total = TILES_M * (128 / 16);
        gemm_wmma_f32<96, 128, true><<<ceil_div(total, GB / 32), GB, 0, stream>>>(bufB, W2, bufA, total);
    }
    agg_init<128><<<ceil_div((long long)NN * 128, TB), TB, 0, stream>>>(bufA, dis, b2, bufB);
    agg_scatter<128><<<ceil_div((long long)E * 128, TB), TB, 0, stream>>>(bufA, row, col, norm, bufB, E);

    // ---- layer 3: 128 -> 64 ------------------------------------------------
    {
        int total = TILES_M * (64 / 16);
        gemm_wmma_f32<128, 64, false><<<ceil_div(total, GB / 32), GB, 0, stream>>>(bufB, W3, bufA, total);
    }
    agg_init<64><<<ceil_div((long long)NN * 64, TB), TB, 0, stream>>>(bufA, dis, b3, bufB);
    agg_scatter<64><<<ceil_div((long long)E * 64, TB), TB, 0, stream>>>(bufA, row, col, norm, bufB, E);

    // ---- layer 4: 64 -> 32 (straight into d_out) ---------------------------
    {
        int total = TILES_M * (32 / 16);
        gemm_wmma_f32<64, 32, false><<<ceil_div(total, GB / 32), GB, 0, stream>>>(bufB, W4, bufA, total);
    }
    float* out = (float*)d_out;
    agg_init<32><<<ceil_div((long long)NN * 32, TB), TB, 0, stream>>>(bufA, dis, b4, out);
    agg_scatter<32><<<ceil_div((long long)E * 32, TB), TB, 0, stream>>>(bufA, row, col, norm, out, E);
}